// RIS_SDMA_Precoding_26645977105041
// MI455X (gfx1250) — compile-verified
//
#include <hip/hip_runtime.h>
#include <math.h>

#define NC 32
#define NT 64
#define KU 2
#define BATCH 4096
#define PAIRS (BATCH * NC)   // 131072 rows through the MLP
#define MLP_WAVES 2          // 2 waves per block -> 32 KB static LDS

typedef float v2f __attribute__((ext_vector_type(2)));
typedef float v8f __attribute__((ext_vector_type(8)));

// ---------------- complex helpers (float2-backed) ----------------
struct c32 { float re, im; };
__device__ __forceinline__ c32 cadd(c32 a, c32 b) { return {a.re + b.re, a.im + b.im}; }
__device__ __forceinline__ c32 csub(c32 a, c32 b) { return {a.re - b.re, a.im - b.im}; }
__device__ __forceinline__ c32 cmul(c32 a, c32 b) { return {a.re * b.re - a.im * b.im, a.re * b.im + a.im * b.re}; }
__device__ __forceinline__ c32 cconj(c32 a) { return {a.re, -a.im}; }
__device__ __forceinline__ c32 cscale(c32 a, float s) { return {a.re * s, a.im * s}; }
__device__ __forceinline__ float cabs2(c32 a) { return a.re * a.re + a.im * a.im; }
__device__ __forceinline__ c32 ldc(const float2* p, int i) { float2 v = p[i]; return {v.x, v.y}; }
__device__ __forceinline__ void stc(float2* p, int i, c32 v) { p[i] = make_float2(v.re, v.im); }

// ---------------- kernel 0: analog precoder F_RF[b,t,k] ----------------
// F_RF = conj(H1[b,k,Nc/2,t]) / |.| / sqrt(Nt);  H1 imag = H[...,0]
__global__ void k_frf(const float* __restrict__ H, float2* __restrict__ FRF) {
  int idx = blockIdx.x * blockDim.x + threadIdx.x;   // BATCH*NT*KU
  if (idx >= BATCH * NT * KU) return;
  int k = idx & 1;
  int t = (idx >> 1) & (NT - 1);
  int b = idx >> 7;
  const float* hrow = H + (((size_t)b * KU + k) * NC + NC / 2) * (2 * NT);
  float hre = hrow[t];
  float him = hrow[0];
  float mag = sqrtf(hre * hre + him * him);
  float s = 1.0f / (mag * 8.0f);                     // 8 = sqrt(Nt)
  FRF[idx] = make_float2(hre * s, -him * s);
}

// ---------------- kernel 1: equalized channel + MLP features ----------------
// H_equ[b2,n2,i,j] = sum_t H1[b2, n2/16, (n2%16)*2+i, t] * F_RF[(b2%128)*32+n2, t, j]
__global__ void k_feat(const float* __restrict__ H, const float2* __restrict__ FRF,
                       float2* __restrict__ Hc, float* __restrict__ X) {
  int r = blockIdx.x * blockDim.x + threadIdx.x;     // pair = b2*32+n2
  if (r >= PAIRS) return;
  int b2 = r >> 5, n2 = r & 31;
  int kh = n2 >> 4;
  int chb = (n2 & 15) * 2;
  int fb = (b2 & 127) * 32 + n2;

  c32 acc[2][2];
  for (int i = 0; i < 2; ++i) for (int j = 0; j < 2; ++j) acc[i][j] = {0.f, 0.f};

  for (int i = 0; i < 2; ++i) {
    const float* hrow = H + (((size_t)b2 * KU + kh) * NC + (chb + i)) * (2 * NT);
    float him = hrow[0];
    for (int t = 0; t < NT; ++t) {
      c32 h = {hrow[t], him};
      c32 f0 = ldc(FRF, (fb * NT + t) * 2 + 0);
      c32 f1 = ldc(FRF, (fb * NT + t) * 2 + 1);
      acc[i][0] = cadd(acc[i][0], cmul(h, f0));
      acc[i][1] = cadd(acc[i][1], cmul(h, f1));
    }
  }
  float pw = 0.f;
  for (int i = 0; i < 2; ++i) for (int j = 0; j < 2; ++j) pw += cabs2(acc[i][j]);
  float inv = 1.0f / sqrtf(pw);
  for (int i = 0; i < 2; ++i)
    for (int j = 0; j < 2; ++j) {
      c32 v = cscale(acc[i][j], inv);
      stc(Hc, r * 4 + i * 2 + j, v);
      X[(size_t)r * 8 + i * 2 + j] = v.re;
      X[(size_t)r * 8 + 4 + i * 2 + j] = v.im;
    }
}

// ---------------- kernel 2: fused MLP 8->256->256->256->8 via f32 WMMA ----------------
__global__ void __launch_bounds__(MLP_WAVES * 32)
k_mlp(const float* __restrict__ X,
      const float* __restrict__ W0, const float* __restrict__ B0,
      const float* __restrict__ W1, const float* __restrict__ B1,
      const float* __restrict__ W2, const float* __restrict__ B2,
      const float* __restrict__ W3, const float* __restrict__ B3,
      float* __restrict__ Y) {
  __shared__ float lds[MLP_WAVES * 16 * 256];
  const int lane = threadIdx.x & 31;
  const int wave = threadIdx.x >> 5;
  const int rowBase = (blockIdx.x * MLP_WAVES + wave) * 16;
  float* buf = &lds[wave * 16 * 256];
  const int col = lane & 15;        // A row / B-C column index
  const int hiHalf = lane >> 4;     // 0: lanes 0-15, 1: lanes 16-31
  const int kofs = hiHalf * 2;      // lanes 16-31 carry K second half

  v8f acc[16];

  // ---- layer 1: K=8 ----
  #pragma unroll
  for (int j = 0; j < 16; ++j) {
    float bb = B0[j * 16 + col];
    #pragma unroll
    for (int rr = 0; rr < 8; ++rr) acc[j][rr] = bb;
  }
  for (int k0 = 0; k0 < 8; k0 += 4) {
    v2f a;
    a.x = X[(size_t)(rowBase + col) * 8 + k0 + kofs];
    a.y = X[(size_t)(rowBase + col) * 8 + k0 + kofs + 1];
    #pragma unroll
    for (int j = 0; j < 16; ++j) {
      v2f bv;
      bv.x = W0[(k0 + kofs) * 256 + j * 16 + col];
      bv.y = W0[(k0 + kofs + 1) * 256 + j * 16 + col];
      acc[j] = __builtin_amdgcn_wmma_f32_16x16x4_f32(false, a, false, bv, (short)0, acc[j], false, false);
    }
  }
  #pragma unroll
  for (int j = 0; j < 16; ++j)
    #pragma unroll
    for (int rr = 0; rr < 8; ++rr)
      buf[(rr + hiHalf * 8) * 256 + j * 16 + col] = fmaxf(acc[j][rr], 0.0f);
  __syncthreads();

  // ---- layers 2,3: K=256 ----
  for (int layer = 0; layer < 2; ++layer) {
    const float* W = layer ? W2 : W1;
    const float* Bv = layer ? B2 : B1;
    #pragma unroll
    for (int j = 0; j < 16; ++j) {
      float bb = Bv[j * 16 + col];
      #pragma unroll
      for (int rr = 0; rr < 8; ++rr) acc[j][rr] = bb;
    }
    for (int k0 = 0; k0 < 256; k0 += 4) {
      v2f a;
      a.x = buf[col * 256 + k0 + kofs];
      a.y = buf[col * 256 + k0 + kofs + 1];
      #pragma unroll
      for (int j = 0; j < 16; ++j) {
        v2f bv;
        bv.x = W[(k0 + kofs) * 256 + j * 16 + col];
        bv.y = W[(k0 + kofs + 1) * 256 + j * 16 + col];
        acc[j] = __builtin_amdgcn_wmma_f32_16x16x4_f32(false, a, false, bv, (short)0, acc[j], false, false);
      }
    }
    __syncthreads();
    #pragma unroll
    for (int j = 0; j < 16; ++j)
      #pragma unroll
      for (int rr = 0; rr < 8; ++rr)
        buf[(rr + hiHalf * 8) * 256 + j * 16 + col] = fmaxf(acc[j][rr], 0.0f);
    __syncthreads();
  }

  // ---- layer 4: 256 -> 8 (B columns 8..15 zero-padded) ----
  v8f a4;
  {
    float bb = (col < 8) ? B3[col] : 0.0f;
    #pragma unroll
    for (int rr = 0; rr < 8; ++rr) a4[rr] = bb;
  }
  for (int k0 = 0; k0 < 256; k0 += 4) {
    v2f a;
    a.x = buf[col * 256 + k0 + kofs];
    a.y = buf[col * 256 + k0 + kofs + 1];
    v2f bv;
    bv.x = (col < 8) ? W3[(k0 + kofs) * 8 + col] : 0.0f;
    bv.y = (col < 8) ? W3[(k0 + kofs + 1) * 8 + col] : 0.0f;
    a4 = __builtin_amdgcn_wmma_f32_16x16x4_f32(false, a, false, bv, (short)0, a4, false, false);
  }
  if (col < 8) {
    #pragma unroll
    for (int rr = 0; rr < 8; ++rr)
      Y[(size_t)(rowBase + rr + hiHalf * 8) * 8 + col] = a4[rr];
  }
}

// ---------------- kernel 3: per-pair B_base (2x2 complex) and UW ----------------
__global__ void k_prep(const float* __restrict__ Y, const float2* __restrict__ Hc,
                       float2* __restrict__ Bb, float2* __restrict__ UWa) {
  int r = blockIdx.x * blockDim.x + threadIdx.x;
  if (r >= PAIRS) return;
  const float* y = Y + (size_t)r * 8;
  c32 Bij[2][2];
  for (int i = 0; i < 2; ++i) for (int j = 0; j < 2; ++j) Bij[i][j] = {0.f, 0.f};
  for (int k = 0; k < 2; ++k) {
    c32 U = {y[k], y[2 + k]};
    c32 Wm = {y[4 + k], y[6 + k]};
    c32 coef = cscale(Wm, cabs2(U));        // Wm * |U|^2
    stc(UWa, r * 2 + k, cmul(U, Wm));       // U * Wm
    c32 h0 = ldc(Hc, r * 4 + k * 2 + 0);
    c32 h1 = ldc(Hc, r * 4 + k * 2 + 1);
    c32 c0 = cconj(h0), c1 = cconj(h1);
    Bij[0][0] = cadd(Bij[0][0], cmul(coef, cmul(c0, h0)));
    Bij[0][1] = cadd(Bij[0][1], cmul(coef, cmul(c0, h1)));
    Bij[1][0] = cadd(Bij[1][0], cmul(coef, cmul(c1, h0)));
    Bij[1][1] = cadd(Bij[1][1], cmul(coef, cmul(c1, h1)));
  }
  for (int i = 0; i < 2; ++i)
    for (int j = 0; j < 2; ++j) stc(Bb, r * 4 + i * 2 + j, Bij[i][j]);
}

// 2x2 complex inverse of (B + K*mu*I)
__device__ __forceinline__ void binv2(const float2* Bb, int r, float mu, c32 Binv[2][2]) {
  c32 B00 = ldc(Bb, r * 4 + 0), B01 = ldc(Bb, r * 4 + 1);
  c32 B10 = ldc(Bb, r * 4 + 2), B11 = ldc(Bb, r * 4 + 3);
  float d = 2.0f * mu;                      // K * mu
  B00.re += d; B11.re += d;
  c32 det = csub(cmul(B00, B11), cmul(B01, B10));
  float id = 1.0f / cabs2(det);
  c32 idet = {det.re * id, -det.im * id};
  Binv[0][0] = cmul(B11, idet);
  Binv[0][1] = cscale(cmul(B01, idet), -1.0f);
  Binv[1][0] = cscale(cmul(B10, idet), -1.0f);
  Binv[1][1] = cmul(B00, idet);
}

// ---------------- bisection: scalar init ----------------
__global__ void k_init(float* scal) {
  if (threadIdx.x == 0 && blockIdx.x == 0) { scal[0] = 0.0f; scal[1] = 10.0f; scal[2] = 5.0f; }
}

// ---------------- bisection: Pitem partial sums (deterministic tree) ----------------
__global__ void k_pitem(const float2* __restrict__ Bb, const float2* __restrict__ UWa,
                        const float2* __restrict__ Hc, const float* __restrict__ scal,
                        float* __restrict__ partial) {
  __shared__ float red[256];
  int r = blockIdx.x * 256 + threadIdx.x;   // grid covers PAIRS exactly
  float mu = 0.5f * (scal[0] + scal[1]);
  float p = 0.0f;
  c32 Binv[2][2];
  binv2(Bb, r, mu, Binv);
  for (int k = 0; k < 2; ++k) {
    c32 c0 = cconj(ldc(Hc, r * 4 + k * 2 + 0));
    c32 c1 = cconj(ldc(Hc, r * 4 + k * 2 + 1));
    c32 g0 = cadd(cmul(Binv[0][0], c0), cmul(Binv[0][1], c1));
    c32 g1 = cadd(cmul(Binv[1][0], c0), cmul(Binv[1][1], c1));
    p += cabs2(ldc(UWa, r * 2 + k)) * (cabs2(g0) + cabs2(g1));
  }
  red[threadIdx.x] = p;
  __syncthreads();
  for (int s = 128; s > 0; s >>= 1) {
    if ((int)threadIdx.x < s) red[threadIdx.x] += red[threadIdx.x + s];
    __syncthreads();
  }
  if (threadIdx.x == 0) partial[blockIdx.x] = red[0];
}

// ---------------- bisection: scalar update ----------------
__global__ void k_update(float* scal, const float* __restrict__ partial) {
  if (threadIdx.x != 0 || blockIdx.x != 0) return;
  float sum = 0.0f;
  for (int i = 0; i < 512; ++i) sum += partial[i];
  float mu = 0.5f * (scal[0] + scal[1]);
  if (sum > 1.0f) scal[0] = mu; else scal[1] = mu;
  scal[2] = mu;
}

// ---------------- final V with converged mu ----------------
__global__ void k_v(const float2* __restrict__ Bb, const float2* __restrict__ UWa,
                    const float2* __restrict__ Hc, const float* __restrict__ scal,
                    float2* __restrict__ Va) {
  int r = blockIdx.x * blockDim.x + threadIdx.x;
  if (r >= PAIRS) return;
  float mu = scal[2];
  c32 Binv[2][2];
  binv2(Bb, r, mu, Binv);
  for (int k = 0; k < 2; ++k) {
    c32 c0 = cconj(ldc(Hc, r * 4 + k * 2 + 0));
    c32 c1 = cconj(ldc(Hc, r * 4 + k * 2 + 1));
    c32 g0 = cadd(cmul(Binv[0][0], c0), cmul(Binv[0][1], c1));
    c32 g1 = cadd(cmul(Binv[1][0], c0), cmul(Binv[1][1], c1));
    c32 uw = ldc(UWa, r * 2 + k);
    stc(Va, r * 4 + 0 * 2 + k, cmul(uw, g0));  // V[i=0][k]
    stc(Va, r * 4 + 1 * 2 + k, cmul(uw, g1));  // V[i=1][k]
  }
}

// ---------------- PowS[n,b] = sum_{t,k} |F_RF[b] @ F_BB[n,b]|^2 ----------------
// F_BB flat reinterpret: G[n,b] = V_flat[n*BATCH + b]
__global__ void k_pows(const float2* __restrict__ Va, const float2* __restrict__ FRF,
                       float* __restrict__ PowS) {
  int q = blockIdx.x * blockDim.x + threadIdx.x;   // q = n*BATCH + b
  if (q >= PAIRS) return;
  int b = q & (BATCH - 1);
  c32 G00 = ldc(Va, q * 4 + 0), G01 = ldc(Va, q * 4 + 1);
  c32 G10 = ldc(Va, q * 4 + 2), G11 = ldc(Va, q * 4 + 3);
  float pw = 0.0f;
  for (int t = 0; t < NT; ++t) {
    c32 f0 = ldc(FRF, (b * NT + t) * 2 + 0);
    c32 f1 = ldc(FRF, (b * NT + t) * 2 + 1);
    c32 s0 = cadd(cmul(f0, G00), cmul(f1, G10));
    c32 s1 = cadd(cmul(f0, G01), cmul(f1, G11));
    pw += cabs2(s0) + cabs2(s1);
  }
  PowS[q] = pw;
}

// ---------------- final normalized F_SDMA -> output ----------------
__global__ void k_out(const float2* __restrict__ Va, const float2* __restrict__ FRF,
                      const float* __restrict__ PowS, float* __restrict__ out) {
  int idx = blockIdx.x * blockDim.x + threadIdx.x; // BATCH*NC*NT
  if (idx >= BATCH * NC * NT) return;
  int t = idx & (NT - 1);
  int n2 = (idx >> 6) & (NC - 1);
  int b2 = idx >> 11;
  int q = n2 * BATCH + b2;
  float Pw = PowS[(b2 >> 7) * BATCH + (b2 & 127) * 32 + n2];
  float s = sqrtf(2.0f) / sqrtf(Pw);               // sqrt(K)/sqrt(Power)
  c32 G00 = cscale(ldc(Va, q * 4 + 0), s), G01 = cscale(ldc(Va, q * 4 + 1), s);
  c32 G10 = cscale(ldc(Va, q * 4 + 2), s), G11 = cscale(ldc(Va, q * 4 + 3), s);
  c32 f0 = ldc(FRF, (b2 * NT + t) * 2 + 0);
  c32 f1 = ldc(FRF, (b2 * NT + t) * 2 + 1);
  c32 s0 = cadd(cmul(f0, G00), cmul(f1, G10));
  c32 s1 = cadd(cmul(f0, G01), cmul(f1, G11));
  float* orow = out + (size_t)b2 * 8192;
  int cb = n2 * 128 + t * 2;
  orow[cb + 0] = s0.re;
  orow[cb + 1] = s1.re;
  orow[4096 + cb + 0] = s0.im;
  orow[4096 + cb + 1] = s1.im;
}

extern "C" void kernel_launch(void* const* d_in, const int* in_sizes, int n_in,
                              void* d_out, int out_size, void* d_ws, size_t ws_size,
                              hipStream_t stream) {
  const float* H  = (const float*)d_in[0];
  const float* W0 = (const float*)d_in[1]; const float* B0 = (const float*)d_in[2];
  const float* W1 = (const float*)d_in[3]; const float* B1 = (const float*)d_in[4];
  const float* W2 = (const float*)d_in[5]; const float* B2 = (const float*)d_in[6];
  const float* W3 = (const float*)d_in[7]; const float* B3 = (const float*)d_in[8];
  float* out = (float*)d_out;

  char* ws = (char*)d_ws;
  size_t o = 0;
  float2* FRF = (float2*)(ws + o); o += (size_t)BATCH * NT * KU * sizeof(float2); // 4 MB
  float2* Hc  = (float2*)(ws + o); o += (size_t)PAIRS * 4 * sizeof(float2);       // 4 MB
  float*  X   = (float*) (ws + o); o += (size_t)PAIRS * 8 * sizeof(float);        // 4 MB
  float*  Yv  = (float*) (ws + o); o += (size_t)PAIRS * 8 * sizeof(float);        // 4 MB
  float2* Bb  = (float2*)(ws + o); o += (size_t)PAIRS * 4 * sizeof(float2);       // 4 MB
  float2* UWa = (float2*)(ws + o); o += (size_t)PAIRS * 2 * sizeof(float2);       // 2 MB
  float2* Va  = (float2*)(ws + o); o += (size_t)PAIRS * 4 * sizeof(float2);       // 4 MB
  float*  PowS= (float*) (ws + o); o += (size_t)PAIRS * sizeof(float);            // 0.5 MB
  float*  part= (float*) (ws + o); o += 512 * sizeof(float);
  float*  scal= (float*) (ws + o); o += 16;

  k_frf <<<(BATCH * NT * KU) / 256, 256, 0, stream>>>(H, FRF);
  k_feat<<<PAIRS / 256, 256, 0, stream>>>(H, FRF, Hc, X);
  k_mlp <<<PAIRS / (16 * MLP_WAVES), MLP_WAVES * 32, 0, stream>>>(
      X, W0, B0, W1, B1, W2, B2, W3, B3, Yv);
  k_prep<<<PAIRS / 256, 256, 0, stream>>>(Yv, Hc, Bb, UWa);
  k_init<<<1, 1, 0, stream>>>(scal);
  // hi-lo = 10*2^-t exactly in f32 -> loop runs exactly 20 iterations
  for (int it = 0; it < 20; ++it) {
    k_pitem<<<PAIRS / 256, 256, 0, stream>>>(Bb, UWa, Hc, scal, part);
    k_update<<<1, 1, 0, stream>>>(scal, part);
  }
  k_v   <<<PAIRS / 256, 256, 0, stream>>>(Bb, UWa, Hc, scal, Va);
  k_pows<<<PAIRS / 256, 256, 0, stream>>>(Va, FRF, PowS);
  k_out <<<(BATCH * NC * NT) / 256, 256, 0, stream>>>(Va, FRF, PowS, out);
}